// SGC_17394617548983
// MI455X (gfx1250) — compile-verified
//
#include <hip/hip_runtime.h>

typedef float v2f __attribute__((ext_vector_type(2)));
typedef float v8f __attribute__((ext_vector_type(8)));

#define N_NODES   100000
#define N_EDGES   1000000
#define D_FEAT    80
#define N_CLASSES 40
#define NPAD      48          // 40 cols padded to 3 WMMA n-tiles
#define KPAIRS    (D_FEAT/2)  // 40 float2 k-pairs

// ---------------------------------------------------------------------------
// 1) degree counting: atomicAdd int per edge endpoint
// ---------------------------------------------------------------------------
__global__ void degree_count_kernel(const int* __restrict__ S, const int* __restrict__ R,
                                    int* cnt_s, int* cnt_r, int nE) {
    int e = blockIdx.x * blockDim.x + threadIdx.x;
    if (e < nE) {
        atomicAdd(&cnt_s[S[e]], 1);
        atomicAdd(&cnt_r[R[e]], 1);
    }
}

// ---------------------------------------------------------------------------
// 2) counts -> rsqrt(count+1) in place (self edge => deg>=1 => max(deg,1)=deg)
// ---------------------------------------------------------------------------
__global__ void inv_sqrt_kernel(int* cs, int* cr, int n) {
    int i = blockIdx.x * blockDim.x + threadIdx.x;
    if (i < n) {
        float a = rsqrtf((float)(cs[i] + 1));
        float b = rsqrtf((float)(cr[i] + 1));
        ((float*)cs)[i] = a;
        ((float*)cr)[i] = b;
    }
}

// ---------------------------------------------------------------------------
// 3) Y = diag(inv_s) * (X @ W) via V_WMMA_F32_16X16X4_F32
//    8 waves/block, one 16-row M-tile per wave, 3 N-tiles (48 padded cols).
//    W staged in LDS as k-pair float2: Wl[p*NPAD + c] = {W[2p][c], W[2p+1][c]}
// ---------------------------------------------------------------------------
__global__ void __launch_bounds__(256)
gemm_xw_scale_kernel(const float* __restrict__ X, const float* __restrict__ W,
                     const float* __restrict__ inv_s, float* __restrict__ Y, int nNodes) {
    __shared__ v2f Wl[KPAIRS * NPAD];   // 40*48*8B = 15360 B LDS

    for (int i = threadIdx.x; i < KPAIRS * NPAD; i += blockDim.x) {
        int p = i / NPAD, c = i - p * NPAD;
        v2f w = {0.f, 0.f};
        if (c < N_CLASSES) {
            w.x = W[(2 * p    ) * N_CLASSES + c];
            w.y = W[(2 * p + 1) * N_CLASSES + c];
        }
        Wl[i] = w;
    }
    __syncthreads();

    const int wave = threadIdx.x >> 5;
    const int lane = threadIdx.x & 31;
    const int half = lane >> 4;     // 0: K=k0..k0+1, 1: K=k0+2..k0+3
    const int lr   = lane & 15;

    const int row0 = (blockIdx.x * 8 + wave) * 16;
    if (row0 >= nNodes) return;     // wave-uniform: EXEC stays all-1s for WMMA

    const v2f* xrow = (const v2f*)(X + (size_t)(row0 + lr) * D_FEAT); // A row, lane lr
    v8f acc0 = {}, acc1 = {}, acc2 = {};

#pragma unroll
    for (int k0 = 0; k0 < D_FEAT; k0 += 4) {
        const int p = (k0 >> 1) + half;
        v2f a  = xrow[p];                    // A[lr][k0+2*half .. +1]
        v2f b0 = Wl[p * NPAD + lr];          // B[k][n-tile 0]
        v2f b1 = Wl[p * NPAD + 16 + lr];     // B[k][n-tile 1]
        v2f b2 = Wl[p * NPAD + 32 + lr];     // B[k][n-tile 2] (zero-padded)
        acc0 = __builtin_amdgcn_wmma_f32_16x16x4_f32(false, a, false, b0, (short)0, acc0, false, false);
        acc1 = __builtin_amdgcn_wmma_f32_16x16x4_f32(false, a, false, b1, (short)0, acc1, false, false);
        acc2 = __builtin_amdgcn_wmma_f32_16x16x4_f32(false, a, false, b2, (short)0, acc2, false, false);
    }

    // C/D layout: VGPR j, lanes 0-15 -> M=j, N=lane; lanes 16-31 -> M=j+8, N=lane-16
#pragma unroll
    for (int j = 0; j < 8; ++j) {
        int row = row0 + j + 8 * half;       // row0+15 <= 99999: always in range
        float sc = inv_s[row];
        float* yrow = Y + (size_t)row * N_CLASSES;
        yrow[lr]      = acc0[j] * sc;
        yrow[16 + lr] = acc1[j] * sc;
        if (32 + lr < N_CLASSES) yrow[32 + lr] = acc2[j] * sc;
    }
}

// ---------------------------------------------------------------------------
// 4) edge scatter: dst[r] += src[s], 10 threads/edge, float4 gather + f32 atomics
// ---------------------------------------------------------------------------
__global__ void edge_scatter_kernel(const float* __restrict__ src, float* dst,
                                    const int* __restrict__ S, const int* __restrict__ R,
                                    int nE) {
    int tid = blockIdx.x * blockDim.x + threadIdx.x;
    int e = tid / 10;
    int q = tid - e * 10;
    if (e >= nE) return;
    int s = S[e], r = R[e];
    float4 v = *(const float4*)(src + (size_t)s * N_CLASSES + q * 4);
    float* d = dst + (size_t)r * N_CLASSES + q * 4;
    atomicAdd(d + 0, v.x);
    atomicAdd(d + 1, v.y);
    atomicAdd(d + 2, v.z);
    atomicAdd(d + 3, v.w);
}

// ---------------------------------------------------------------------------
// 5) Z = (U + Y) * inv_r * inv_s   (self edge + layer-1 post / layer-2 pre scale)
//    written in place over Y
// ---------------------------------------------------------------------------
__global__ void scale_mid_kernel(const float* __restrict__ U, float* YZ,
                                 const float* __restrict__ inv_r,
                                 const float* __restrict__ inv_s, int n) {
    int tid = blockIdx.x * blockDim.x + threadIdx.x;
    int i = tid / 10;
    int q = tid - i * 10;
    if (i >= n) return;
    float sc = inv_r[i] * inv_s[i];
    size_t off = (size_t)i * N_CLASSES + q * 4;
    float4 u = *(const float4*)(U + off);
    float4 y = *(const float4*)(YZ + off);
    float4 z = make_float4((u.x + y.x) * sc, (u.y + y.y) * sc,
                           (u.z + y.z) * sc, (u.w + y.w) * sc);
    *(float4*)(YZ + off) = z;
}

// ---------------------------------------------------------------------------
// 6) O = (O + Z) * inv_r  (self edge + final scale), in place
// ---------------------------------------------------------------------------
__global__ void final_scale_kernel(float* O, const float* __restrict__ Z,
                                   const float* __restrict__ inv_r, int n) {
    int tid = blockIdx.x * blockDim.x + threadIdx.x;
    int i = tid / 10;
    int q = tid - i * 10;
    if (i >= n) return;
    float sc = inv_r[i];
    size_t off = (size_t)i * N_CLASSES + q * 4;
    float4 o = *(const float4*)(O + off);
    float4 z = *(const float4*)(Z + off);
    float4 r = make_float4((o.x + z.x) * sc, (o.y + z.y) * sc,
                           (o.z + z.z) * sc, (o.w + z.w) * sc);
    *(float4*)(O + off) = r;
}

// ---------------------------------------------------------------------------
extern "C" void kernel_launch(void* const* d_in, const int* in_sizes, int n_in,
                              void* d_out, int out_size, void* d_ws, size_t ws_size,
                              hipStream_t stream) {
    const float* X = (const float*)d_in[0];   // [N,80]
    const int*   S = (const int*)d_in[1];     // [E]
    const int*   R = (const int*)d_in[2];     // [E]
    const float* W = (const float*)d_in[3];   // [80,40]
    float*       O = (float*)d_out;           // [N,40]

    char* ws = (char*)d_ws;
    const size_t ALN = 256;
    auto up = [&](size_t x) { return (x + ALN - 1) & ~(ALN - 1); };
    size_t offS = 0;
    size_t offR = offS + up((size_t)N_NODES * 4);
    size_t offY = offR + up((size_t)N_NODES * 4);
    size_t offU = offY + up((size_t)N_NODES * N_CLASSES * 4);

    int*   cnt_s = (int*)(ws + offS);
    int*   cnt_r = (int*)(ws + offR);
    float* inv_s = (float*)(ws + offS);   // overwritten in place by inv_sqrt_kernel
    float* inv_r = (float*)(ws + offR);
    float* Y     = (float*)(ws + offY);   // Y, then Z in place
    float* U     = (float*)(ws + offU);

    // zero accumulators (graph-capturable memset nodes)
    hipMemsetAsync(cnt_s, 0, (size_t)N_NODES * 4, stream);
    hipMemsetAsync(cnt_r, 0, (size_t)N_NODES * 4, stream);
    hipMemsetAsync(U,     0, (size_t)N_NODES * N_CLASSES * 4, stream);
    hipMemsetAsync(O,     0, (size_t)N_NODES * N_CLASSES * 4, stream);

    degree_count_kernel<<<(N_EDGES + 255) / 256, 256, 0, stream>>>(S, R, cnt_s, cnt_r, N_EDGES);
    inv_sqrt_kernel<<<(N_NODES + 255) / 256, 256, 0, stream>>>(cnt_s, cnt_r, N_NODES);

    int mtiles = (N_NODES + 15) / 16;                 // 6250
    int gblocks = (mtiles + 7) / 8;                   // 782
    gemm_xw_scale_kernel<<<gblocks, 256, 0, stream>>>(X, W, inv_s, Y, N_NODES);

    int escatter_blocks = ((size_t)N_EDGES * 10 + 255) / 256;
    int nscale_blocks   = ((size_t)N_NODES * 10 + 255) / 256;

    edge_scatter_kernel<<<escatter_blocks, 256, 0, stream>>>(Y, U, S, R, N_EDGES);
    scale_mid_kernel<<<nscale_blocks, 256, 0, stream>>>(U, Y, inv_r, inv_s, N_NODES);
    edge_scatter_kernel<<<escatter_blocks, 256, 0, stream>>>(Y, O, S, R, N_EDGES);
    final_scale_kernel<<<nscale_blocks, 256, 0, stream>>>(O, Y, inv_r, N_NODES);
}